// WaveletTransform2D_89996744720684
// MI455X (gfx1250) — compile-verified
//
#include <hip/hip_runtime.h>
#include <cstdint>

// ---------------------------------------------------------------------------
// 2D Haar wavelet (8,256,256,256) fp32 -> 4 x (8,256,128,128), memory-bound.
// CDNA5 path: double-buffered GLOBAL_LOAD_ASYNC_TO_LDS_B128 staging with
// s_wait_asynccnt, conflict-free ds_load_b128 reads, non-temporal b128 stores.
// Roofline: 1 GiB total traffic @ 23.3 TB/s -> ~46 us floor; compute is free.
// ---------------------------------------------------------------------------

typedef float v4f __attribute__((ext_vector_type(4)));
typedef int   v4i __attribute__((ext_vector_type(4)));

#define THREADS          256
#define TILE_ROWS        16                      // input rows per tile
#define UNITS_PER_ROW    64                      // 256 floats / 4 per row
#define UNITS_PER_TILE   (TILE_ROWS * UNITS_PER_ROW)   // 1024 float4 = 16 KiB
#define LOADS_PER_THREAD (UNITS_PER_TILE / THREADS)    // 4
#define TILES_PER_IMAGE  (256 / TILE_ROWS)             // 16

// ---- CDNA5 async global->LDS copy (16B per lane), guarded -----------------
#if defined(__has_builtin)
#  if __has_builtin(__builtin_amdgcn_global_load_async_to_lds_b128)
#    define HAVE_ASYNC_LDS 1
#  endif
#endif
#ifndef HAVE_ASYNC_LDS
#  define HAVE_ASYNC_LDS 0
#endif

__device__ __forceinline__ void async_copy16(const v4f* g, v4f* l) {
#if HAVE_ASYNC_LDS
  // This toolchain declares the builtin with generic int4* parameters
  // (per the round-1 diagnostic); it handles address spaces in CodeGen.
  __builtin_amdgcn_global_load_async_to_lds_b128(
      (v4i*)(void*)g, (v4i*)(void*)l, 0, 0);
#else
  *l = *g;   // synchronous fallback (compile-probe path)
#endif
}

#if HAVE_ASYNC_LDS && defined(__has_builtin) && __has_builtin(__builtin_amdgcn_s_wait_asynccnt)
#  define WAIT_ASYNC(n) __builtin_amdgcn_s_wait_asynccnt(n)
#elif HAVE_ASYNC_LDS
#  define WAIT_ASYNC(n) asm volatile("s_wait_asynccnt %0" :: "i"(n) : "memory")
#else
#  define WAIT_ASYNC(n) ((void)0)
#endif

// Even/odd de-interleave swizzle for the LDS destination: float4 column c4 of
// a row goes to slot c4/2 (even) or 32 + c4/2 (odd). Compute threads then read
// contiguous 512B per wave per ds_load_b128 -> zero bank conflicts.
__device__ __forceinline__ int lds_slot(int u) {
  const int row = u >> 6;
  const int c4  = u & 63;
  return row * UNITS_PER_ROW + ((c4 & 1) ? (32 + (c4 >> 1)) : (c4 >> 1));
}

__global__ __launch_bounds__(THREADS)
void WaveletTransform2D_haar_kernel(const float* __restrict__ xin,
                                    float* __restrict__ out,
                                    int numTiles,
                                    long long sbStride4 /* float4 per subband */) {
  __shared__ v4f tile[2][UNITS_PER_TILE];          // 32 KiB double buffer

  const int  t    = threadIdx.x;
  const v4f* gx   = (const v4f*)xin;
  v4f*       gout = (v4f*)out;

  int tileIdx = (int)blockIdx.x;
  if (tileIdx >= numTiles) return;
  const int stride = (int)gridDim.x;

  // ---- prefetch tile 0 into buffer 0 (4 async b128 ops per wave) ----
  {
    const int n  = tileIdx / TILES_PER_IMAGE;
    const int tr = tileIdx % TILES_PER_IMAGE;
    const long long base = (long long)n * (256LL * 256 / 4)
                         + (long long)tr * UNITS_PER_TILE;
#pragma unroll
    for (int k = 0; k < LOADS_PER_THREAD; ++k) {
      const int u = t + k * THREADS;               // lanes contiguous -> coalesced
      async_copy16(gx + base + u, &tile[0][lds_slot(u)]);
    }
  }

  int buf = 0;
  for (; tileIdx < numTiles; tileIdx += stride) {
    // ---- prefetch next tile into the other buffer, then wait for current ----
    const int nextTile = tileIdx + stride;
    if (nextTile < numTiles) {
      const int n  = nextTile / TILES_PER_IMAGE;
      const int tr = nextTile % TILES_PER_IMAGE;
      const long long base = (long long)n * (256LL * 256 / 4)
                           + (long long)tr * UNITS_PER_TILE;
#pragma unroll
      for (int k = 0; k < LOADS_PER_THREAD; ++k) {
        const int u = t + k * THREADS;
        async_copy16(gx + base + u, &tile[buf ^ 1][lds_slot(u)]);
      }
      WAIT_ASYNC(LOADS_PER_THREAD);   // newest 4 in flight; current tile landed
    } else {
      WAIT_ASYNC(0);                  // drain
    }
    __syncthreads();                  // cross-wave LDS visibility

    // ---- compute: thread owns out-row r (within tile), out cols 4gq..4gq+3 --
    const int r  = t >> 5;            // 0..7  (constant per wave)
    const int gq = t & 31;            // lane  -> contiguous LDS/global access
    const v4f r0e = tile[buf][(2 * r)     * UNITS_PER_ROW + gq];       // row 2r  cols 8gq..+3
    const v4f r0o = tile[buf][(2 * r)     * UNITS_PER_ROW + 32 + gq];  // row 2r  cols 8gq+4..+7
    const v4f r1e = tile[buf][(2 * r + 1) * UNITS_PER_ROW + gq];
    const v4f r1o = tile[buf][(2 * r + 1) * UNITS_PER_ROW + 32 + gq];

    v4f LL, LH, HL, HH;
    {
      const float a = r0e.x, b = r0e.y, c = r1e.x, d = r1e.y;
      LL.x = (a + b + c + d) * 0.5f;  LH.x = (-a + b - c + d) * 0.5f;
      HL.x = (-a - b + c + d) * 0.5f; HH.x = (a - b - c + d) * 0.5f;
    }
    {
      const float a = r0e.z, b = r0e.w, c = r1e.z, d = r1e.w;
      LL.y = (a + b + c + d) * 0.5f;  LH.y = (-a + b - c + d) * 0.5f;
      HL.y = (-a - b + c + d) * 0.5f; HH.y = (a - b - c + d) * 0.5f;
    }
    {
      const float a = r0o.x, b = r0o.y, c = r1o.x, d = r1o.y;
      LL.z = (a + b + c + d) * 0.5f;  LH.z = (-a + b - c + d) * 0.5f;
      HL.z = (-a - b + c + d) * 0.5f; HH.z = (a - b - c + d) * 0.5f;
    }
    {
      const float a = r0o.z, b = r0o.w, c = r1o.z, d = r1o.w;
      LL.w = (a + b + c + d) * 0.5f;  LH.w = (-a + b - c + d) * 0.5f;
      HL.w = (-a - b + c + d) * 0.5f; HH.w = (a - b - c + d) * 0.5f;
    }

    // ---- stores: one float4 per subband, NT (stream doesn't fit in L2) ----
    const int n  = tileIdx / TILES_PER_IMAGE;
    const int tr = tileIdx % TILES_PER_IMAGE;
    const long long o4 = (long long)n * 4096                       // 128*128/4 per image
                       + (long long)((TILE_ROWS / 2) * tr + r) * 32
                       + gq;
    __builtin_nontemporal_store(LL, gout + o4);
    __builtin_nontemporal_store(LH, gout + sbStride4 + o4);
    __builtin_nontemporal_store(HL, gout + 2 * sbStride4 + o4);
    __builtin_nontemporal_store(HH, gout + 3 * sbStride4 + o4);

    __syncthreads();                  // protect buffer reuse next iteration
    buf ^= 1;
  }
}

extern "C" void kernel_launch(void* const* d_in, const int* in_sizes, int n_in,
                              void* d_out, int out_size, void* d_ws, size_t ws_size,
                              hipStream_t stream) {
  (void)n_in; (void)d_ws; (void)ws_size; (void)out_size;
  const float* x   = (const float*)d_in[0];
  float*       out = (float*)d_out;

  const long long total     = (long long)in_sizes[0];         // B*C*H*W
  const int       images    = (int)(total / (256LL * 256LL)); // B*C, H=W=256
  const int       numTiles  = images * TILES_PER_IMAGE;
  const long long sbStride4 = total / 16;                     // float4 per subband

  int blocks = numTiles < 8192 ? numTiles : 8192;             // grid-stride, ~4 tiles/block
  WaveletTransform2D_haar_kernel<<<blocks, THREADS, 0, stream>>>(
      x, out, numTiles, sbStride4);
}